// PDR_89378269430400
// MI455X (gfx1250) — compile-verified
//
#include <hip/hip_runtime.h>
#include <hip/hip_bf16.h>
#include <stdint.h>

#define N_NODES 100000
#define N_EDGES 1600000
#define D       128
#define ROWS    64            // nodes per block in the WMMA kernel
#define LSTRIDE 132           // LDS row stride (floats): conflict-free for b64 reads

typedef __attribute__((ext_vector_type(2))) float v2f;
typedef __attribute__((ext_vector_type(8))) float v8f;

// ---------------------------------------------------------------------------
// 0) zero the scratch region used for accumulation (agg1 [N*D] | cnt [N] | agg2 [N])
// ---------------------------------------------------------------------------
__global__ void PDR_zero(float* __restrict__ p, int n) {
    int i = blockIdx.x * blockDim.x + threadIdx.x;
    if (i < n) p[i] = 0.0f;
}

// ---------------------------------------------------------------------------
// 1) layer-1 edge scatter: one wave32 per edge, float4 per lane (32*4 = 128)
//    agg1[dst,:] += x[src,:];  cnt[dst] += 1
//    x (51MB) and agg1 (51MB) both fit in the 192MB L2 -> L2-speed atomics.
// ---------------------------------------------------------------------------
__global__ void __launch_bounds__(256) PDR_scatter1(
        const float* __restrict__ x,
        const long long* __restrict__ ei,   // [2, E] int64
        float* __restrict__ agg,
        float* __restrict__ cnt) {
    int e = blockIdx.x * 8 + (threadIdx.x >> 5);
    if (e >= N_EDGES) return;
    int lane = threadIdx.x & 31;
    __builtin_prefetch(&ei[e + 512], 0, 1);   // gfx1250 global_prefetch_b8
    long long src = ei[e];
    long long dst = ei[N_EDGES + e];
    float4 v = ((const float4*)(x + (size_t)src * D))[lane];
    float* a = agg + (size_t)dst * D + lane * 4;
    atomicAdd(a + 0, v.x);
    atomicAdd(a + 1, v.y);
    atomicAdd(a + 2, v.z);
    atomicAdd(a + 3, v.w);
    if (lane == 0) atomicAdd(&cnt[dst], 1.0f);
}

// ---------------------------------------------------------------------------
// 2) fused SAGE layer-1 GEMM via V_WMMA_F32_16X16X4_F32 (native fp32 WMMA):
//      h = (agg/max(cnt,1)) @ w_l^T + b_l + x @ w_r^T
//    Block = 256 thr = 8 waves; block owns ROWS=64 nodes x all 128 cols.
//    Wave w owns cols [16w,16w+16) and 4 row-subtiles -> each B (weight) v2f
//    load feeds 4 WMMAs; A operands staged in LDS (mean pre-scaled by 1/cnt).
//    WMMA operand roles (f32, 16x16x4):
//      A: lane l -> row (l&15),  K = kb + (l>>4)*2 + {0,1}   (v2f, LDS b64)
//      B: lane l -> col (l&15),  same K pair                 (v2f, global b64)
//      C: VGPR r, lane l -> row r + (l>>4)*8, col (l&15)
// ---------------------------------------------------------------------------
__global__ void __launch_bounds__(256) PDR_sage1_wmma(
        const float* __restrict__ x,
        const float* __restrict__ agg,
        const float* __restrict__ cnt,
        const float* __restrict__ w_l,   // [128,128]
        const float* __restrict__ b_l,   // [128]
        const float* __restrict__ w_r,   // [128,128]
        float* __restrict__ h) {
    __shared__ float sMean[ROWS * LSTRIDE];
    __shared__ float sX[ROWS * LSTRIDE];
    __shared__ float sRinv[ROWS];

    const int tid  = threadIdx.x;
    const int wave = tid >> 5;
    const int lane = tid & 31;
    const int row0 = blockIdx.x * ROWS;

    // phase 0: per-row 1/max(cnt,1)
    if (tid < ROWS) {
        int r = row0 + tid;
        float c = cnt[min(r, N_NODES - 1)];
        sRinv[tid] = 1.0f / fmaxf(c, 1.0f);
    }
    __syncthreads();

    // phase 1: stage A operands (64 rows x 32 float4 = 2048 float4 per matrix)
    for (int i = tid; i < ROWS * (D / 4); i += 256) {
        int row = i >> 5;                 // 32 float4 per row
        int c4  = i & 31;
        int gr  = min(row0 + row, N_NODES - 1);
        float rv = sRinv[row];
        float4 a = ((const float4*)(agg + (size_t)gr * D))[c4];
        a.x *= rv; a.y *= rv; a.z *= rv; a.w *= rv;
        *(float4*)&sMean[row * LSTRIDE + c4 * 4] = a;
        float4 xv = ((const float4*)(x + (size_t)gr * D))[c4];
        *(float4*)&sX[row * LSTRIDE + c4 * 4] = xv;
    }
    __syncthreads();

    // phase 2: WMMA compute (no divergence -> EXEC all ones)
    const int m   = lane & 15;
    const int kh  = lane >> 4;
    const int col0 = wave * 16;
    const float* wlRow = w_l + (size_t)(col0 + m) * D;
    const float* wrRow = w_r + (size_t)(col0 + m) * D;
    const float* sM = sMean + m * LSTRIDE;
    const float* sXm = sX   + m * LSTRIDE;

    v8f acc0 = {}, acc1 = {}, acc2 = {}, acc3 = {};
#pragma unroll 4
    for (int kb = 0; kb < D; kb += 4) {
        const int k0 = kb + kh * 2;
        v2f bL = *(const v2f*)(wlRow + k0);
        v2f a0 = *(const v2f*)(sM + 0 * 16 * LSTRIDE + k0);
        v2f a1 = *(const v2f*)(sM + 1 * 16 * LSTRIDE + k0);
        v2f a2 = *(const v2f*)(sM + 2 * 16 * LSTRIDE + k0);
        v2f a3 = *(const v2f*)(sM + 3 * 16 * LSTRIDE + k0);
        acc0 = __builtin_amdgcn_wmma_f32_16x16x4_f32(false, a0, false, bL, (short)0, acc0, false, false);
        acc1 = __builtin_amdgcn_wmma_f32_16x16x4_f32(false, a1, false, bL, (short)0, acc1, false, false);
        acc2 = __builtin_amdgcn_wmma_f32_16x16x4_f32(false, a2, false, bL, (short)0, acc2, false, false);
        acc3 = __builtin_amdgcn_wmma_f32_16x16x4_f32(false, a3, false, bL, (short)0, acc3, false, false);
        v2f bR = *(const v2f*)(wrRow + k0);
        v2f x0 = *(const v2f*)(sXm + 0 * 16 * LSTRIDE + k0);
        v2f x1 = *(const v2f*)(sXm + 1 * 16 * LSTRIDE + k0);
        v2f x2 = *(const v2f*)(sXm + 2 * 16 * LSTRIDE + k0);
        v2f x3 = *(const v2f*)(sXm + 3 * 16 * LSTRIDE + k0);
        acc0 = __builtin_amdgcn_wmma_f32_16x16x4_f32(false, x0, false, bR, (short)0, acc0, false, false);
        acc1 = __builtin_amdgcn_wmma_f32_16x16x4_f32(false, x1, false, bR, (short)0, acc1, false, false);
        acc2 = __builtin_amdgcn_wmma_f32_16x16x4_f32(false, x2, false, bR, (short)0, acc2, false, false);
        acc3 = __builtin_amdgcn_wmma_f32_16x16x4_f32(false, x3, false, bR, (short)0, acc3, false, false);
    }

    // phase 3: add bias, store (guard tail rows only here, after all WMMAs)
    float bias = b_l[col0 + m];
    v8f accs[4] = {acc0, acc1, acc2, acc3};
#pragma unroll
    for (int s = 0; s < 4; ++s) {
#pragma unroll
        for (int r = 0; r < 8; ++r) {
            int mr = row0 + s * 16 + kh * 8 + r;
            if (mr < N_NODES)
                h[(size_t)mr * D + col0 + m] = accs[s][r] + bias;
        }
    }
}

// ---------------------------------------------------------------------------
// 3) layer-2 projection (exploit linearity of mean): per node
//    zl = relu(h)·w2_l , zr = relu(h)·w2_r   — one wave per node
// ---------------------------------------------------------------------------
__global__ void __launch_bounds__(256) PDR_proj2(
        const float* __restrict__ h,
        const float* __restrict__ w2l,   // [128]
        const float* __restrict__ w2r,   // [128]
        float* __restrict__ zl,
        float* __restrict__ zr) {
    int node = blockIdx.x * 8 + (threadIdx.x >> 5);
    if (node >= N_NODES) return;
    int lane = threadIdx.x & 31;
    float4 v  = ((const float4*)(h + (size_t)node * D))[lane];
    v.x = fmaxf(v.x, 0.0f); v.y = fmaxf(v.y, 0.0f);
    v.z = fmaxf(v.z, 0.0f); v.w = fmaxf(v.w, 0.0f);
    float4 wl = ((const float4*)w2l)[lane];
    float4 wr = ((const float4*)w2r)[lane];
    float sl = v.x * wl.x + v.y * wl.y + v.z * wl.z + v.w * wl.w;
    float sr = v.x * wr.x + v.y * wr.y + v.z * wr.z + v.w * wr.w;
#pragma unroll
    for (int off = 16; off > 0; off >>= 1) {
        sl += __shfl_down(sl, off, 32);
        sr += __shfl_down(sr, off, 32);
    }
    if (lane == 0) { zl[node] = sl; zr[node] = sr; }
}

// ---------------------------------------------------------------------------
// 4) layer-2 edge scatter: scalar per edge (128x less traffic than naive)
// ---------------------------------------------------------------------------
__global__ void PDR_scatter2(const long long* __restrict__ ei,
                             const float* __restrict__ zl,
                             float* __restrict__ agg2) {
    int e = blockIdx.x * blockDim.x + threadIdx.x;
    if (e >= N_EDGES) return;
    long long src = ei[e];
    long long dst = ei[N_EDGES + e];
    atomicAdd(&agg2[dst], zl[src]);
}

// ---------------------------------------------------------------------------
// 5) finalize: out = sigmoid(agg2/max(cnt,1) + b2 + zr)
// ---------------------------------------------------------------------------
__global__ void PDR_final(const float* __restrict__ agg2,
                          const float* __restrict__ cnt,
                          const float* __restrict__ zr,
                          const float* __restrict__ b2,
                          float* __restrict__ out) {
    int i = blockIdx.x * blockDim.x + threadIdx.x;
    if (i >= N_NODES) return;
    float s = agg2[i] / fmaxf(cnt[i], 1.0f) + b2[0] + zr[i];
    out[i] = 1.0f / (1.0f + __expf(-s));
}

// ---------------------------------------------------------------------------
extern "C" void kernel_launch(void* const* d_in, const int* in_sizes, int n_in,
                              void* d_out, int out_size, void* d_ws, size_t ws_size,
                              hipStream_t stream) {
    const float*     x    = (const float*)d_in[0];
    const long long* ei   = (const long long*)d_in[1];   // int64 [2,E]
    const float*     w1_l = (const float*)d_in[2];
    const float*     b1_l = (const float*)d_in[3];
    const float*     w1_r = (const float*)d_in[4];
    const float*     w2_l = (const float*)d_in[5];
    const float*     b2_l = (const float*)d_in[6];
    const float*     w2_r = (const float*)d_in[7];

    float* out = (float*)d_out;                 // [N]
    float* h   = (float*)d_out + N_NODES;       // [N,128] (second tuple output)

    // workspace layout (floats): agg1[N*D] | cnt[N] | agg2[N] | zl[N] | zr[N]
    float* agg1 = (float*)d_ws;
    float* cnt  = agg1 + (size_t)N_NODES * D;
    float* agg2 = cnt  + N_NODES;
    float* zl   = agg2 + N_NODES;
    float* zr   = zl   + N_NODES;

    // zero agg1 + cnt + agg2 (contiguous)
    int nz = N_NODES * D + 2 * N_NODES;
    PDR_zero<<<(nz + 255) / 256, 256, 0, stream>>>(agg1, nz);

    // layer-1 aggregation
    PDR_scatter1<<<N_EDGES / 8, 256, 0, stream>>>(x, ei, agg1, cnt);

    // layer-1 fused WMMA GEMM -> h
    PDR_sage1_wmma<<<(N_NODES + ROWS - 1) / ROWS, 256, 0, stream>>>(
        x, agg1, cnt, w1_l, b1_l, w1_r, h);

    // layer-2 projection to scalars
    PDR_proj2<<<(N_NODES + 7) / 8, 256, 0, stream>>>(h, w2_l, w2_r, zl, zr);

    // layer-2 scalar aggregation
    PDR_scatter2<<<(N_EDGES + 255) / 256, 256, 0, stream>>>(ei, zl, agg2);

    // finalize with sigmoid
    PDR_final<<<(N_NODES + 255) / 256, 256, 0, stream>>>(agg2, cnt, zr, b2_l, out);
}